// CstLoss_30408368455835
// MI455X (gfx1250) — compile-verified
//
#include <hip/hip_runtime.h>
#include <math.h>

typedef __attribute__((ext_vector_type(2))) float v2f;
typedef __attribute__((ext_vector_type(8))) float v8f;

#define FLT_NEG_MAX (-3.4028234663852886e+38f)

// ---------------- wave32 reductions ----------------
__device__ inline float wredMax(float v) {
#pragma unroll
  for (int off = 16; off > 0; off >>= 1)
    v = fmaxf(v, __shfl_xor(v, off, 32));
  return v;
}
__device__ inline float wredSum(float v) {
#pragma unroll
  for (int off = 16; off > 0; off >>= 1)
    v += __shfl_xor(v, off, 32);
  return v;
}

// block reduce across 6 waves (192 threads); red must hold >= 6 floats
__device__ inline float bredMax6(float v, float* red, int tid) {
  float w = wredMax(v);
  __syncthreads();                 // protect previous use of red[]
  if ((tid & 31) == 0) red[tid >> 5] = w;
  __syncthreads();
  float r = red[0];
#pragma unroll
  for (int i = 1; i < 6; ++i) r = fmaxf(r, red[i]);
  return r;
}
__device__ inline float bredSum6(float v, float* red, int tid) {
  float w = wredSum(v);
  __syncthreads();
  if ((tid & 31) == 0) red[tid >> 5] = w;
  __syncthreads();
  float r = red[0];
#pragma unroll
  for (int i = 1; i < 6; ++i) r += red[i];
  return r;
}

// softmax over the 192-vector (one element per thread) + l2-normalize, store
__device__ inline void softnorm_store(float x, float* dst, float* red, int tid) {
  float m = bredMax6(x, red, tid);
  float e = expf(x - m);
  float s = bredSum6(e, red, tid);
  float p = e / s;
  float ss = bredSum6(p * p, red, tid);
  float nrm = sqrtf(ss);
  *dst = p / fmaxf(nrm, 1e-8f);
}

// ---------------------------------------------------------------------------
// Kernel 1: per (n,c) 192x192 slice of both tensors:
//   row-max over w -> (192), col-max over h -> (192), for output and target,
//   then fused softmax + L2-normalize of the four vectors -> A arrays in ws.
// Block = 192 threads (6 waves). Inputs read exactly once, coalesced b64.
// ---------------------------------------------------------------------------
__global__ void __launch_bounds__(192) stats_kernel(
    const float* __restrict__ gout, const float* __restrict__ gtgt,
    float* __restrict__ Ao_h, float* __restrict__ At_h,
    float* __restrict__ Ao_w, float* __restrict__ At_w) {
  __shared__ float ro[192], rt[192], co[192], ct[192];
  __shared__ v2f cop[6 * 96], ctp[6 * 96];
  __shared__ float red[8];

  const int tid = threadIdx.x;
  const int lane = tid & 31;
  const int wv = tid >> 5;                // 0..5
  const size_t base = (size_t)blockIdx.x * (192 * 192);
  const v2f* o2 = (const v2f*)(gout + base);
  const v2f* t2 = (const v2f*)(gtgt + base);

  v2f com[3], ctm[3];
#pragma unroll
  for (int j = 0; j < 3; ++j) {
    com[j].x = FLT_NEG_MAX; com[j].y = FLT_NEG_MAX;
    ctm[j].x = FLT_NEG_MAX; ctm[j].y = FLT_NEG_MAX;
  }

  // wave wv owns rows h = wv + 6*r (32 rows); each wave reads full rows.
  for (int r = 0; r < 32; ++r) {
    const int h = wv + 6 * r;
    float rom = FLT_NEG_MAX, rtm = FLT_NEG_MAX;
#pragma unroll
    for (int j = 0; j < 3; ++j) {
      const int c2 = j * 32 + lane;       // float2 column index (0..95)
      v2f vo = o2[h * 96 + c2];
      v2f vt = t2[h * 96 + c2];
      com[j].x = fmaxf(com[j].x, vo.x); com[j].y = fmaxf(com[j].y, vo.y);
      ctm[j].x = fmaxf(ctm[j].x, vt.x); ctm[j].y = fmaxf(ctm[j].y, vt.y);
      rom = fmaxf(rom, fmaxf(vo.x, vo.y));
      rtm = fmaxf(rtm, fmaxf(vt.x, vt.y));
    }
    rom = wredMax(rom);
    rtm = wredMax(rtm);
    if (lane == 0) { ro[h] = rom; rt[h] = rtm; }
  }

  // publish per-wave column-max partials
#pragma unroll
  for (int j = 0; j < 3; ++j) {
    cop[wv * 96 + j * 32 + lane] = com[j];
    ctp[wv * 96 + j * 32 + lane] = ctm[j];
  }
  __syncthreads();

  // combine column partials: thread tid owns column tid
  {
    const float* cf = (const float*)cop;
    const float* tf = (const float*)ctp;
    float m1 = cf[tid], m2 = tf[tid];
#pragma unroll
    for (int w = 1; w < 6; ++w) {
      m1 = fmaxf(m1, cf[w * 192 + tid]);
      m2 = fmaxf(m2, tf[w * 192 + tid]);
    }
    co[tid] = m1; ct[tid] = m2;
  }
  __syncthreads();

  const float x_oh = ro[tid], x_th = rt[tid];
  const float x_ow = co[tid], x_tw = ct[tid];
  const size_t obase = (size_t)blockIdx.x * 192 + tid;

  softnorm_store(x_oh, Ao_h + obase, red, tid);
  softnorm_store(x_th, At_h + obase, red, tid);
  softnorm_store(x_ow, Ao_w + obase, red, tid);
  softnorm_store(x_tw, At_w + obase, red, tid);
}

// ---------------------------------------------------------------------------
// Kernel 2: per (n, direction): S = Ao(64x192) @ At(64x192)^T via
// V_WMMA_F32_16X16X4_F32. sim[n] = sum(S); sim_pos[n,c] = diag(S).
// Block = 128 threads (4 waves); wave w owns i-tile w, loops 4 j-tiles, K=192.
// ---------------------------------------------------------------------------
__global__ void __launch_bounds__(128) pairwise_wmma(
    const float* __restrict__ Ao_h, const float* __restrict__ At_h,
    const float* __restrict__ Ao_w, const float* __restrict__ At_w,
    float* __restrict__ sp_h, float* __restrict__ sp_w,
    float* __restrict__ s_h, float* __restrict__ s_w) {
  const int bx = blockIdx.x;
  const int n = bx >> 1;
  const int dir = bx & 1;
  const float* A = dir ? Ao_w : Ao_h;
  const float* B = dir ? At_w : At_h;
  float* sp = dir ? sp_w : sp_h;
  float* sout = dir ? s_w : s_h;

  const int tid = threadIdx.x;
  const int lane = tid & 31;
  const int wv = tid >> 5;            // i-tile index 0..3
  const int i0 = wv * 16;
  const size_t CK = (size_t)64 * 192;

  // A-matrix 16x4 f32 fragment layout: lanes 0-15 hold (row=M, K=k0..k0+1),
  // lanes 16-31 hold (row=M, K=k0+2..k0+3). B = At^T has identical addressing.
  const float* Abase = A + (size_t)n * CK + (size_t)(i0 + (lane & 15)) * 192 + ((lane >> 4) << 1);
  const float* Bbase = B + (size_t)n * CK + (size_t)(lane & 15) * 192 + ((lane >> 4) << 1);

  v8f acc0 = {}, acc1 = {}, acc2 = {}, acc3 = {};
  for (int k0 = 0; k0 < 192; k0 += 4) {
    v2f a  = *(const v2f*)(Abase + k0);
    v2f b0 = *(const v2f*)(Bbase + k0);
    v2f b1 = *(const v2f*)(Bbase + 16 * 192 + k0);
    v2f b2 = *(const v2f*)(Bbase + 32 * 192 + k0);
    v2f b3 = *(const v2f*)(Bbase + 48 * 192 + k0);
    acc0 = __builtin_amdgcn_wmma_f32_16x16x4_f32(false, a, false, b0, (short)0, acc0, false, false);
    acc1 = __builtin_amdgcn_wmma_f32_16x16x4_f32(false, a, false, b1, (short)0, acc1, false, false);
    acc2 = __builtin_amdgcn_wmma_f32_16x16x4_f32(false, a, false, b2, (short)0, acc2, false, false);
    acc3 = __builtin_amdgcn_wmma_f32_16x16x4_f32(false, a, false, b3, (short)0, acc3, false, false);
  }

  // full pairwise sum -> sim[n, dir]
  float s = 0.f;
#pragma unroll
  for (int r = 0; r < 8; ++r) s += acc0[r] + acc1[r] + acc2[r] + acc3[r];
  s = wredSum(s);
  __shared__ float red[4];
  if (lane == 0) red[wv] = s;
  __syncthreads();
  if (tid == 0) sout[n] = red[0] + red[1] + red[2] + red[3];

  // diagonal: global diag lives in tile (i==j) == tile wv of this wave.
  // C/D f32 16x16 layout: lanes 0-15 -> N=lane, VGPR r -> M=r;
  //                       lanes 16-31 -> N=lane-16, VGPR r -> M=r+8.
  // So diag(M==N): lanes 0-7 at vgpr=lane, lanes 24-31 at vgpr=lane-24.
  v8f accd = acc0;
  if (wv == 1) accd = acc1;
  else if (wv == 2) accd = acc2;
  else if (wv == 3) accd = acc3;

  float d = 0.f;
#pragma unroll
  for (int r = 0; r < 8; ++r) {
    float cv = accd[r];
    bool sel = (lane < 8) ? (r == lane) : (lane >= 24 && r == (lane - 24));
    d = sel ? cv : d;
  }
  if (lane < 8)        sp[n * 64 + i0 + lane] = d;
  else if (lane >= 24) sp[n * 64 + i0 + 8 + (lane - 24)] = d;
}

// ---------------------------------------------------------------------------
// Kernel 3: loss[n,c] = -log( 0.5*(sp_h+sp_w) / (0.5*(s_h+s_w)) ) / (c*n)
// ---------------------------------------------------------------------------
__global__ void __launch_bounds__(256) finalize_kernel(
    const float* __restrict__ sp_h, const float* __restrict__ sp_w,
    const float* __restrict__ s_h, const float* __restrict__ s_w,
    float* __restrict__ out) {
  const int idx = blockIdx.x * 256 + threadIdx.x;
  if (idx >= 32 * 64) return;
  const int n = idx >> 6;
  const float sp = 0.5f * (sp_h[idx] + sp_w[idx]);
  const float s = 0.5f * (s_h[n] + s_w[n]);
  out[idx] = -logf(sp / s) * (1.0f / 2048.0f);
}

// ---------------------------------------------------------------------------
extern "C" void kernel_launch(void* const* d_in, const int* in_sizes, int n_in,
                              void* d_out, int out_size, void* d_ws, size_t ws_size,
                              hipStream_t stream) {
  (void)in_sizes; (void)n_in; (void)out_size; (void)ws_size;
  const float* gout = (const float*)d_in[0];
  const float* gtgt = (const float*)d_in[1];
  float* ws = (float*)d_ws;

  const size_t AN = (size_t)32 * 64 * 192;   // 393216 floats per A array
  float* Ao_h = ws;
  float* At_h = ws + AN;
  float* Ao_w = ws + 2 * AN;
  float* At_w = ws + 3 * AN;
  float* sp_h = ws + 4 * AN;                 // 2048
  float* sp_w = sp_h + 2048;                 // 2048
  float* s_h  = sp_w + 2048;                 // 32
  float* s_w  = s_h + 32;                    // 32

  // Phase 1: fused max-reduction + softmax + L2-normalize (bandwidth bound)
  stats_kernel<<<32 * 64, 192, 0, stream>>>(gout, gtgt, Ao_h, At_h, Ao_w, At_w);

  // Phase 2: WMMA pairwise channel similarity (f32 tensor path)
  pairwise_wmma<<<32 * 2, 128, 0, stream>>>(Ao_h, At_h, Ao_w, At_w,
                                            sp_h, sp_w, s_h, s_w);

  // Phase 3: final loss
  finalize_kernel<<<(32 * 64 + 255) / 256, 256, 0, stream>>>(
      sp_h, sp_w, s_h, s_w, (float*)d_out);
}